// StateAttention_12893491823241
// MI455X (gfx1250) — compile-verified
//
#include <hip/hip_runtime.h>

typedef __attribute__((ext_vector_type(2))) float v2f;
typedef __attribute__((ext_vector_type(8))) float v8f;
typedef __attribute__((ext_vector_type(4))) unsigned int v4u;
typedef __attribute__((ext_vector_type(8))) int v8i;
typedef __attribute__((ext_vector_type(4))) int v4i;

constexpr int Bsz = 2048;
constexpr int Lsz = 128;
constexpr int Dsz = 512;

constexpr int CHUNK_ROWS = 16;                 // K-rows staged per TDM transfer
constexpr int NCHUNK = Lsz / CHUNK_ROWS;       // 8 chunks per batch row
constexpr int SEG_DW = 260;                    // 256 data dwords + 4 pad dwords
constexpr int ROW_DW = 2 * SEG_DW;             // padded row stride = 520 dwords
constexpr int CHUNK_DW = CHUNK_ROWS * ROW_DW;  // 8320 dwords = 33280 B
constexpr int W_OFF = 2 * CHUNK_DW;            // weight row after 2 chunk buffers

// ---------------------------------------------------------------------------
// Kernel 1: new_a_t stencil (tiny).  new_a[b,l] =
//   (l==0)*a[0]*r0 + (l<127)*(1-pm[l+1])*ws[l] + (l>0)*pm[l]*ws[l-1]
// with ws[l] = a[l]*r1 + a[l+1]*r0.
// ---------------------------------------------------------------------------
__global__ __launch_bounds__(Lsz) void new_at_kernel(
    const float* __restrict__ a_t, const float* __restrict__ r_t,
    const float* __restrict__ pm, float* __restrict__ new_a) {
  const int b = blockIdx.x;
  const int l = threadIdx.x;
  const float r0 = r_t[2 * b + 0];
  const float r1 = r_t[2 * b + 1];
  const float* a = a_t + (size_t)b * Lsz;
  const float* m = pm + (size_t)b * Lsz;

  const float a_l = a[l];
  float v = 0.0f;
  if (l == 0) v += a_l * r0;
  if (l < Lsz - 1) v += (1.0f - m[l + 1]) * (a_l * r1 + a[l + 1] * r0);
  if (l > 0) v += m[l] * (a[l - 1] * r1 + a_l * r0);
  new_a[(size_t)b * Lsz + l] = v;
}

// ---------------------------------------------------------------------------
// TDM descriptor issue: move one 16x512 f32 chunk (row-major, row stride 512
// elements) from global memory into LDS at lds_byte_off, padding 4 dwords
// after every 256 dwords stored (-> 520-dword LDS row stride, which makes the
// WMMA B-operand reads bank-conflict-free: lanes 16-31 read row r+2, i.e.
// 1040 dwords = +16 banks away from lanes 0-15).
//
// D# per CDNA5 ISA ch.8:
//  group0: [1:0]count=1 | [63:32]lds_addr | [120:64]global_addr | [127:126]type=2
//  group1: data_size=4B, pad_enable=1, pad_interval=256dw(code7),
//          pad_amount=4dw(code3), tensor_dim0=512, tensor_dim1=16,
//          tile_dim0=512, tile_dim1=16, tensor_dim0_stride=512
//  group2/3: zero (2D tensor); trailing int32x8 + cpol per the 6-arg
//  clang-23 builtin signature (zero-filled trailing operand).
// ---------------------------------------------------------------------------
__device__ __forceinline__ void tdm_load_chunk(unsigned long long gaddr,
                                               unsigned int lds_byte_off) {
  v4u g0;
  g0[0] = 1u;                                        // count=1 (valid)
  g0[1] = lds_byte_off;                              // LDS byte address
  g0[2] = (unsigned int)(gaddr & 0xFFFFFFFFull);     // global_addr[31:0]
  g0[3] = (unsigned int)((gaddr >> 32) & 0x1FFFFFFull) | (2u << 30); // +type=2

  v8i g1;
  g1[0] = (int)0x07D20000u;   // mask=0 | data_size=2(4B)<<16 | pad_en<<20 |
                              // pad_interval=7(256dw)<<22 | pad_amount=3(4dw)<<25
  g1[1] = (int)(512u << 16);  // tensor_dim0[15:0]=512 in bits[63:48]
  g1[2] = (int)(16u << 16);   // tensor_dim0[31:16]=0, tensor_dim1=16
  g1[3] = (int)(512u << 16);  // tensor_dim1[31:16]=0, tile_dim0=512
  g1[4] = 16;                 // tile_dim1=16, tile_dim2=0
  g1[5] = 512;                // tensor_dim0_stride[31:0]=512
  g1[6] = 0;                  // stride[47:32]=0, tensor_dim1_stride[15:0]=0
  g1[7] = 0;

  v4i g2 = {0, 0, 0, 0};
  v4i g3 = {0, 0, 0, 0};
  v8i g4 = {0, 0, 0, 0, 0, 0, 0, 0};  // trailing operand (6-arg toolchain)
  __builtin_amdgcn_tensor_load_to_lds(g0, g1, g2, g3, g4, 0);
}

// ---------------------------------------------------------------------------
// Kernel 2: output[b,d] = sum_l w[b,l] * E[b,l,d].
// One block (8 waves) per batch row.  TDM double-buffers 16x512 chunks of E
// into LDS (TENSORcnt pipeline, wave 0 drives the DMA); all waves consume the
// staged chunk with V_WMMA_F32_16X16X4_F32, 4 d-tiles per wave, A-rows
// replicated (only M=0 of C is stored; FLOP waste irrelevant at 0.5 FLOP/B).
//
// VGPR layouts (ISA 7.12.2):
//   A 16x4 f32 : lanes 0-15 -> K=0,1 ; lanes 16-31 -> K=2,3   (per-lane v2f)
//   B 4x16 f32 : VGPR v, lanes 0-15 -> K=v ; lanes 16-31 -> K=v+2, N=lane%16
//   C 16x16 f32: VGPR0, lanes 0-15 -> M=0, N=lane
// ---------------------------------------------------------------------------
__global__ __launch_bounds__(256) void einsum_tdm_wmma_kernel(
    const float* __restrict__ new_a, const float* __restrict__ emb,
    float* __restrict__ out) {
  // Single shared array => LDS offset 0: [buf0 | buf1 | weight row]
  __shared__ float smem[2 * CHUNK_DW + Lsz];

  const int b = blockIdx.x;
  const int tid = threadIdx.x;
  const int wave = tid >> 5;
  const int lane = tid & 31;
  const int n = lane & 15;             // B/C column within a 16-wide tile
  const int half2 = (lane >> 4) << 1;  // 0 (lanes 0-15) or 2 (lanes 16-31)

  // Stage the weight row into the LDS tail.
  if (tid < Lsz) smem[W_OFF + tid] = new_a[(size_t)b * Lsz + tid];

  const unsigned long long ebase =
      (unsigned long long)(const void*)(emb + (size_t)b * Lsz * Dsz);
  constexpr unsigned int CHUNK_GBYTES = CHUNK_ROWS * Dsz * 4;  // 32 KB
  constexpr unsigned int CHUNK_LBYTES = CHUNK_DW * 4;          // 33280 B

  if (wave == 0) tdm_load_chunk(ebase, 0u);  // prime the pipeline (chunk 0)

  v8f c0 = {}, c1 = {}, c2 = {}, c3 = {};

  for (int k = 0; k < NCHUNK; ++k) {
    if (wave == 0) {
      if (k + 1 < NCHUNK) {
        tdm_load_chunk(ebase + (unsigned long long)(k + 1) * CHUNK_GBYTES,
                       ((unsigned)(k + 1) & 1u) * CHUNK_LBYTES);
        __builtin_amdgcn_s_wait_tensorcnt(1);  // chunk k finished
      } else {
        __builtin_amdgcn_s_wait_tensorcnt(0);  // last chunk finished
      }
    }
    __syncthreads();  // chunk k visible to all waves

    const int base = (k & 1) ? CHUNK_DW : 0;

    // A-operand pairs for the 4 K-steps of this chunk (broadcast LDS reads).
    v2f av[4];
#pragma unroll
    for (int kk = 0; kk < 4; ++kk) {
      const int row = 4 * kk + half2;
      av[kk].x = smem[W_OFF + k * CHUNK_ROWS + row];
      av[kk].y = smem[W_OFF + k * CHUNK_ROWS + row + 1];
    }

    v8f* acc[4] = {&c0, &c1, &c2, &c3};
#pragma unroll
    for (int t = 0; t < 4; ++t) {
      const int d0 = (wave + t * 8) * 16;  // this wave's d-tile
      const int seg_add = d0 >> 8;         // column-class (pad segment select)
      const int col = (d0 & 255) + n;
      v8f c = *acc[t];
#pragma unroll
      for (int kk = 0; kk < 4; ++kk) {
        const int row = 4 * kk + half2;
        const int idx = (row * 2 + seg_add) * SEG_DW + col;
        v2f bv;
        bv.x = smem[base + idx];           // row r   , col d0+n
        bv.y = smem[base + idx + ROW_DW];  // row r+1 , col d0+n
        c = __builtin_amdgcn_wmma_f32_16x16x4_f32(
            /*neg_a=*/false, av[kk], /*neg_b=*/false, bv,
            /*c_mod=*/(short)0, c, /*reuse_a=*/false, /*reuse_b=*/false);
      }
      *acc[t] = c;
    }
    __syncthreads();  // everyone done reading buf[k&1] before it is reloaded
  }

  // Row M=0 of each accumulator holds the 16 output columns (lanes 0-15).
  if (lane < 16) {
    out[(size_t)b * Dsz + (wave + 0 * 8) * 16 + lane] = c0[0];
    out[(size_t)b * Dsz + (wave + 1 * 8) * 16 + lane] = c1[0];
    out[(size_t)b * Dsz + (wave + 2 * 8) * 16 + lane] = c2[0];
    out[(size_t)b * Dsz + (wave + 3 * 8) * 16 + lane] = c3[0];
  }
}

// ---------------------------------------------------------------------------
// Host-side launcher.  d_in order: a_t, r_t, input_embedding, padded_mask, step.
// d_out = [output (B*D) | new_a_t (B*L)] (tuple return order, flattened).
// ---------------------------------------------------------------------------
extern "C" void kernel_launch(void* const* d_in, const int* in_sizes, int n_in,
                              void* d_out, int out_size, void* d_ws,
                              size_t ws_size, hipStream_t stream) {
  const float* a_t = (const float*)d_in[0];
  const float* r_t = (const float*)d_in[1];
  const float* emb = (const float*)d_in[2];
  const float* pmask = (const float*)d_in[3];
  // d_in[4] (step) is unused by the computation.

  float* out = (float*)d_out;              // B*D floats
  float* new_a = out + (size_t)Bsz * Dsz;  // B*L floats

  new_at_kernel<<<Bsz, Lsz, 0, stream>>>(a_t, r_t, pmask, new_a);
  einsum_tdm_wmma_kernel<<<Bsz, 256, 0, stream>>>(new_a, emb, out);
}